// KMIC_29703993820000
// MI455X (gfx1250) — compile-verified
//
#include <hip/hip_runtime.h>
#include <hip/hip_bf16.h>
#include <math.h>

// ---------------------------------------------------------------------------
// Model dims (fixed by the reference)
// ---------------------------------------------------------------------------
#define Bn 64
#define Sn 50
#define Dn 64
#define Hn 64
#define NNn 16
#define RS (Bn * Sn)          // 3200 (b,s) rows

typedef __attribute__((ext_vector_type(2))) float v2f;
typedef __attribute__((ext_vector_type(8))) float v8f;

// ---------------------------------------------------------------------------
// WMMA helpers: fp32 16x16 tile, K accumulated in steps of 4 via
// v_wmma_f32_16x16x4_f32 (CDNA5, wave32).
//
// A (16x4 f32): lanes 0-15 -> M=lane, VGPR0=K0,VGPR1=K1; lanes 16-31 -> K2,K3
// B (4x16 f32): lanes 0-15 -> N=lane, VGPR0=K0,VGPR1=K1; lanes 16-31 -> K2,K3
// C/D (16x16):  lane = (M>=8?16:0)+N ; VGPR v holds row M = v + (lane>=16?8:0)
// ---------------------------------------------------------------------------
__device__ __forceinline__ v8f wmma_k4(v2f a, v2f b, v8f c) {
  return __builtin_amdgcn_wmma_f32_16x16x4_f32(false, a, false, b, (short)0, c,
                                               false, false);
}

// C += A[16xK] * B[KxN-tile], K=64. B stored row-major K x ldB, tile at n0.
__device__ __forceinline__ v8f tile_gemm64(const float* A, int ldA,
                                           const float* Bm, int ldB, int n0,
                                           v8f c) {
  const int lane = threadIdx.x & 31;
  const int half = (lane < 16) ? 0 : 2;
  const int rc = lane & 15;
#pragma unroll
  for (int kk = 0; kk < 16; ++kk) {
    const int k = kk * 4 + half;
    v2f a, b;
    a.x = A[rc * ldA + k];
    a.y = A[rc * ldA + k + 1];
    b.x = Bm[k * ldB + n0 + rc];
    b.y = Bm[(k + 1) * ldB + n0 + rc];
    c = wmma_k4(a, b, c);
  }
  return c;
}

// Same but B is given transposed: B(K=k, N=n) = Bt[n*ldBt + k]
// (matches reference weights stored as W[out,in]).
__device__ __forceinline__ v8f tile_gemm64_bt(const float* A, int ldA,
                                              const float* Bt, int ldBt, int n0,
                                              v8f c) {
  const int lane = threadIdx.x & 31;
  const int half = (lane < 16) ? 0 : 2;
  const int rc = lane & 15;
#pragma unroll
  for (int kk = 0; kk < 16; ++kk) {
    const int k = kk * 4 + half;
    v2f a, b;
    a.x = A[rc * ldA + k];
    a.y = A[rc * ldA + k + 1];
    b.x = Bt[(n0 + rc) * ldBt + k];
    b.y = Bt[(n0 + rc) * ldBt + k + 1];
    c = wmma_k4(a, b, c);
  }
  return c;
}

__device__ __forceinline__ float sigmoidf_(float x) {
  return 1.0f / (1.0f + __expf(-x));
}

// ---------------------------------------------------------------------------
// K0: full KG aggregation per (b,s). One 256-thread block per (b,s).
// Produces x = relu(tanh-hop output) + pos  -> [3200,64]
// ---------------------------------------------------------------------------
__global__ __launch_bounds__(256) void k0_kgat(
    const int* __restrict__ u, const int* __restrict__ click_seq,
    const int* __restrict__ adj_ent, const int* __restrict__ adj_rel,
    const float* __restrict__ usr_emb, const float* __restrict__ ent_emb,
    const float* __restrict__ rel_emb, const float* __restrict__ agg_W,
    const float* __restrict__ agg_b, const float* __restrict__ pos,
    float* __restrict__ x_out) {
  __shared__ float s_user[64];
  __shared__ float s_rel[40 * 64];
  __shared__ float s_W[64 * 64];
  __shared__ float s_b[64];
  __shared__ int s_e0;
  __shared__ int s_e1[16], s_r0[16], s_e2[256], s_r1[256];
  __shared__ float s_l1[256], s_l0[16], s_att1[256], s_att0[16];
  __shared__ float s_M1[16 * 64];
  __shared__ float s_ev1p[16 * 64];
  __shared__ float s_row2a[64], s_ev0p[64], s_row3[64];

  const int t = threadIdx.x;
  const int bs = blockIdx.x;  // b*Sn + s
  const int b = bs / Sn;
  const int s = bs % Sn;
  const int ub = u[b];

  for (int i = t; i < 64; i += 256) s_user[i] = usr_emb[(size_t)ub * 64 + i];
  for (int i = t; i < 40 * 64; i += 256) s_rel[i] = rel_emb[i];
  for (int i = t; i < 64 * 64; i += 256) s_W[i] = agg_W[i];
  if (t < 64) s_b[t] = agg_b[t];
  if (t == 0) s_e0 = click_seq[(size_t)ub * Sn + s];
  __syncthreads();

  const int e0 = s_e0;
  if (t < 16) {
    s_e1[t] = adj_ent[(size_t)e0 * NNn + t];
    s_r0[t] = adj_rel[(size_t)e0 * NNn + t];
  }
  __syncthreads();
  {
    const int j = t >> 4, k = t & 15;
    const int e1 = s_e1[j];
    s_e2[t] = adj_ent[(size_t)e1 * NNn + k];
    s_r1[t] = adj_rel[(size_t)e1 * NNn + k];
  }
  __syncthreads();

  // attention logits: user . rel_emb[r]
  {
    float acc = 0.f;
    const float* rl = s_rel + s_r1[t] * 64;
    for (int d = 0; d < 64; ++d) acc += s_user[d] * rl[d];
    s_l1[t] = acc;
    if (t < 16) {
      float a0 = 0.f;
      const float* r0p = s_rel + s_r0[t] * 64;
      for (int d = 0; d < 64; ++d) a0 += s_user[d] * r0p[d];
      s_l0[t] = a0;
    }
  }
  __syncthreads();

  // softmax over the 16 neighbors of each node
  {
    const int j = t >> 4;
    const float* row = s_l1 + j * 16;
    float m = row[0];
    for (int k = 1; k < 16; ++k) m = fmaxf(m, row[k]);
    float sum = 0.f;
    for (int k = 0; k < 16; ++k) sum += __expf(row[k] - m);
    s_att1[t] = __expf(s_l1[t] - m) / sum;
    if (t < 16) {
      float m0 = s_l0[0];
      for (int k = 1; k < 16; ++k) m0 = fmaxf(m0, s_l0[k]);
      float s0 = 0.f;
      for (int k = 0; k < 16; ++k) s0 += __expf(s_l0[k] - m0);
      s_att0[t] = __expf(s_l0[t] - m0) / s0;
    }
  }
  __syncthreads();

  // hop-1 pre-matmul rows: M1[j] = ev1[j] + sum_k att1[j,k] * ev2[j,k]
  {
    const int j = t >> 4, q = t & 15;
    const int d0 = q * 4;
    float4 acc = *(const float4*)(ent_emb + (size_t)s_e1[j] * 64 + d0);
    for (int k = 0; k < 16; ++k) {
      const float w = s_att1[j * 16 + k];
      const float4 nb =
          *(const float4*)(ent_emb + (size_t)s_e2[j * 16 + k] * 64 + d0);
      acc.x += w * nb.x;
      acc.y += w * nb.y;
      acc.z += w * nb.z;
      acc.w += w * nb.w;
    }
    s_M1[j * 64 + d0 + 0] = acc.x;
    s_M1[j * 64 + d0 + 1] = acc.y;
    s_M1[j * 64 + d0 + 2] = acc.z;
    s_M1[j * 64 + d0 + 3] = acc.w;
  }
  __syncthreads();

  // ev1' = sigmoid(M1 @ agg_W + b)  (WMMA, waves 0..3 = 4 N-tiles)
  {
    const int wv = t >> 5;
    if (wv < 4) {
      v8f c = {};
      c = tile_gemm64(s_M1, 64, s_W, 64, wv * 16, c);
      const int lane = t & 31;
      const int col = wv * 16 + (lane & 15);
      const int rb = (lane < 16) ? 0 : 8;
#pragma unroll
      for (int v = 0; v < 8; ++v)
        s_ev1p[(rb + v) * 64 + col] = sigmoidf_(c[v] + s_b[col]);
    }
  }
  __syncthreads();

  // hop-0, iter 0: row2a = ev0 + sum_j att0[j] * ev1[j]
  if (t < 64) {
    float acc = ent_emb[(size_t)e0 * 64 + t];
    for (int j = 0; j < 16; ++j)
      acc += s_att0[j] * ent_emb[(size_t)s_e1[j] * 64 + t];
    s_row2a[t] = acc;
  }
  __syncthreads();
  if (t < 64) {
    float acc = s_b[t];
    for (int g = 0; g < 64; ++g) acc += s_row2a[g] * s_W[g * 64 + t];
    s_ev0p[t] = sigmoidf_(acc);
  }
  __syncthreads();
  // hop-0, iter 1: row3 = ev0' + sum_j att0[j] * ev1'[j]
  if (t < 64) {
    float acc = s_ev0p[t];
    for (int j = 0; j < 16; ++j) acc += s_att0[j] * s_ev1p[j * 64 + t];
    s_row3[t] = acc;
  }
  __syncthreads();
  if (t < 64) {
    float acc = s_b[t];
    for (int g = 0; g < 64; ++g) acc += s_row3[g] * s_W[g * 64 + t];
    const float item = fmaxf(tanhf(acc), 0.f);
    x_out[(size_t)bs * 64 + t] = item + pos[s * 64 + t];
  }
}

// ---------------------------------------------------------------------------
// K1/K3: batched row GEMM  Y[r] = [relu](X[r] @ W) [+ resid[r]]
// W row-major [64,64]. Grid: 200 blocks x 128 threads (4 waves = 4 N-tiles).
// ---------------------------------------------------------------------------
__global__ __launch_bounds__(128) void k_rowgemm64(
    const float* __restrict__ X, const float* __restrict__ W,
    const float* __restrict__ resid, float* __restrict__ Y, int relu_flag) {
  __shared__ float sX[16 * 64];
  __shared__ float sW[64 * 64];
  const int t = threadIdx.x;
  const size_t r0 = (size_t)blockIdx.x * 16;
  for (int i = t; i < 16 * 64; i += 128) sX[i] = X[r0 * 64 + i];
  for (int i = t; i < 64 * 64; i += 128) sW[i] = W[i];
  __syncthreads();
  const int wv = t >> 5, lane = t & 31;
  v8f c = {};
  c = tile_gemm64(sX, 64, sW, 64, wv * 16, c);
  const int col = wv * 16 + (lane & 15);
  const int rb = (lane < 16) ? 0 : 8;
#pragma unroll
  for (int v = 0; v < 8; ++v) {
    const size_t r = r0 + rb + v;
    float val = c[v];
    if (relu_flag) val = fmaxf(val, 0.f);
    if (resid) val += resid[r * 64 + col];
    Y[r * 64 + col] = val;
  }
}

// ---------------------------------------------------------------------------
// K2: single-head attention per batch row b. 256 threads (8 waves).
//   scores = q k^T / 8 ; masked softmax over 50 ; av = a @ v
// q,k staged in LDS (zero padded to 64 rows); v read from global in av phase
// (a columns >=50 are zero so padded-K contributions vanish).
// ---------------------------------------------------------------------------
__global__ __launch_bounds__(256) void k_attn(const float* __restrict__ q,
                                              const float* __restrict__ k,
                                              const float* __restrict__ v,
                                              float* __restrict__ av) {
  __shared__ float sQ[64 * 64];
  __shared__ float sK[64 * 64];
  __shared__ float sA[64 * 64];
  const int t = threadIdx.x;
  const int b = blockIdx.x;
  const size_t base = (size_t)b * Sn * 64;
  for (int i = t; i < 64 * 64; i += 256) {
    const bool ok = (i >> 6) < Sn;
    sQ[i] = ok ? q[base + i] : 0.f;
    sK[i] = ok ? k[base + i] : 0.f;
  }
  __syncthreads();
  const int wv = t >> 5, lane = t & 31;
  // scores (16 tiles over 8 waves)
  for (int tile = wv; tile < 16; tile += 8) {
    const int mt = tile >> 2, nt = tile & 3;
    v8f c = {};
    c = tile_gemm64_bt(sQ + mt * 16 * 64, 64, sK, 64, nt * 16, c);
    const int col = nt * 16 + (lane & 15);
    const int rb = mt * 16 + ((lane < 16) ? 0 : 8);
#pragma unroll
    for (int vv = 0; vv < 8; ++vv) sA[(rb + vv) * 64 + col] = c[vv] * 0.125f;
  }
  __syncthreads();
  // masked softmax along columns (<50), zero pad rest
  if (t < 64) {
    if (t < Sn) {
      float m = -1e30f;
      for (int n = 0; n < Sn; ++n) m = fmaxf(m, sA[t * 64 + n]);
      float sum = 0.f;
      for (int n = 0; n < Sn; ++n) sum += __expf(sA[t * 64 + n] - m);
      const float inv = 1.f / sum;
      for (int n = 0; n < 64; ++n)
        sA[t * 64 + n] = (n < Sn) ? __expf(sA[t * 64 + n] - m) * inv : 0.f;
    } else {
      for (int n = 0; n < 64; ++n) sA[t * 64 + n] = 0.f;
    }
  }
  __syncthreads();
  // av = A @ V  (V straight from global; K rows >=50 multiplied by zero)
  for (int tile = wv; tile < 16; tile += 8) {
    const int mt = tile >> 2, nt = tile & 3;
    v8f c = {};
    c = tile_gemm64(sA + mt * 16 * 64, 64, v + base, 64, nt * 16, c);
    const int col = nt * 16 + (lane & 15);
    const int rb = mt * 16 + ((lane < 16) ? 0 : 8);
#pragma unroll
    for (int vv = 0; vv < 8; ++vv) {
      const int r = rb + vv;
      if (r < Sn) av[base + (size_t)r * 64 + col] = c[vv];
    }
  }
}

// ---------------------------------------------------------------------------
// K4: xg = trans @ Wx^T + bx   (Wx stored [192,64] -> implicit transpose)
// Grid (200, 3): y picks 64-wide output chunk. 128 threads.
// ---------------------------------------------------------------------------
__global__ __launch_bounds__(128) void k_xg(const float* __restrict__ X,
                                            const float* __restrict__ Wx,
                                            const float* __restrict__ bx,
                                            float* __restrict__ XG) {
  __shared__ float sX[16 * 64];
  const int t = threadIdx.x;
  const size_t r0 = (size_t)blockIdx.x * 16;
  const int g0 = blockIdx.y * 64;
  for (int i = t; i < 16 * 64; i += 128) sX[i] = X[r0 * 64 + i];
  __syncthreads();
  const int wv = t >> 5, lane = t & 31;
  v8f c = {};
  c = tile_gemm64_bt(sX, 64, Wx + (size_t)g0 * 64, 64, wv * 16, c);
  const int col = wv * 16 + (lane & 15);
  const int rb = (lane < 16) ? 0 : 8;
#pragma unroll
  for (int v = 0; v < 8; ++v) {
    const size_t r = r0 + rb + v;
    XG[r * 192 + g0 + col] = c[v] + bx[g0 + col];
  }
}

// ---------------------------------------------------------------------------
// K5: GRU scan. Grid 4 blocks (16 batch rows each), 128 threads (4 waves).
// Per step: hg = h @ Wh^T + bh (12 WMMA N-tiles), gates in VALU.
// Accumulates sum_s relu(h) in registers -> gsum [64,64].
// ---------------------------------------------------------------------------
__global__ __launch_bounds__(128) void k_gru(const float* __restrict__ XG,
                                             const float* __restrict__ Wh,
                                             const float* __restrict__ bh,
                                             int backward,
                                             float* __restrict__ gsum) {
  __shared__ float sh[16 * 64];
  __shared__ float shg[16 * 192];
  const int t = threadIdx.x;
  const int mt = blockIdx.x;  // batch tile
  for (int i = t; i < 16 * 64; i += 128) sh[i] = 0.f;
  float acc[8];
#pragma unroll
  for (int i = 0; i < 8; ++i) acc[i] = 0.f;
  __syncthreads();
  const int wv = t >> 5, lane = t & 31;
  for (int step = 0; step < Sn; ++step) {
    const int s = backward ? (Sn - 1 - step) : step;
    for (int nt = wv * 3; nt < wv * 3 + 3; ++nt) {
      v8f c = {};
      c = tile_gemm64_bt(sh, 64, Wh, 64, nt * 16, c);
      const int col = nt * 16 + (lane & 15);
      const int rb = (lane < 16) ? 0 : 8;
#pragma unroll
      for (int v = 0; v < 8; ++v) shg[(rb + v) * 192 + col] = c[v] + bh[col];
    }
    __syncthreads();
#pragma unroll
    for (int i = 0; i < 8; ++i) {
      const int e = t + i * 128;
      const int row = e >> 6, hc = e & 63;
      const int bb = mt * 16 + row;
      const size_t xb = ((size_t)bb * Sn + s) * 192;
      const float xr = XG[xb + hc];
      const float xz = XG[xb + 64 + hc];
      const float xn = XG[xb + 128 + hc];
      const float hr = shg[row * 192 + hc];
      const float hz = shg[row * 192 + 64 + hc];
      const float hn = shg[row * 192 + 128 + hc];
      const float hprev = sh[e];
      const float r = sigmoidf_(xr + hr);
      const float z = sigmoidf_(xz + hz);
      const float n = tanhf(xn + r * hn);
      const float hnew = (1.f - z) * n + z * hprev;
      sh[e] = hnew;
      acc[i] += fmaxf(hnew, 0.f);
    }
    __syncthreads();
  }
#pragma unroll
  for (int i = 0; i < 8; ++i) {
    const int e = t + i * 128;
    gsum[(size_t)mt * 1024 + e] = acc[i];
  }
}

// ---------------------------------------------------------------------------
// K6: feat = relu([sum relu(gf) | sum relu(gb) | sum trans] @ fc_W + fc_b)
//     out = sigmoid(feat . ent_emb[v])
// ---------------------------------------------------------------------------
__global__ __launch_bounds__(64) void k_final(
    const float* __restrict__ gsf, const float* __restrict__ gsb,
    const float* __restrict__ trans, const float* __restrict__ fc_W,
    const float* __restrict__ fc_b, const float* __restrict__ ent_emb,
    const int* __restrict__ vidx, float* __restrict__ out) {
  __shared__ float fin[192];
  __shared__ float red[64];
  const int t = threadIdx.x;
  const int b = blockIdx.x;
  fin[t] = gsf[b * 64 + t];
  fin[64 + t] = gsb[b * 64 + t];
  float ts = 0.f;
  for (int s = 0; s < Sn; ++s) ts += trans[((size_t)b * Sn + s) * 64 + t];
  fin[128 + t] = ts;
  __syncthreads();
  float acc = fc_b[t];
  for (int g = 0; g < 192; ++g) acc += fin[g] * fc_W[g * 64 + t];
  red[t] = fmaxf(acc, 0.f) * ent_emb[(size_t)vidx[b] * 64 + t];
  __syncthreads();
  if (t == 0) {
    float sum = 0.f;
    for (int d = 0; d < 64; ++d) sum += red[d];
    out[b] = sigmoidf_(sum);
  }
}

// ---------------------------------------------------------------------------
extern "C" void kernel_launch(void* const* d_in, const int* in_sizes, int n_in,
                              void* d_out, int out_size, void* d_ws,
                              size_t ws_size, hipStream_t stream) {
  const int* u = (const int*)d_in[0];
  const int* v = (const int*)d_in[1];
  const int* click_seq = (const int*)d_in[2];
  const int* adj_ent = (const int*)d_in[3];
  const int* adj_rel = (const int*)d_in[4];
  const float* usr_emb = (const float*)d_in[5];
  const float* ent_emb = (const float*)d_in[6];
  const float* rel_emb = (const float*)d_in[7];
  const float* agg_W = (const float*)d_in[8];
  const float* agg_b = (const float*)d_in[9];
  const float* pos = (const float*)d_in[10];
  const float* Wq = (const float*)d_in[11];
  const float* Wk = (const float*)d_in[12];
  const float* Wv = (const float*)d_in[13];
  const float* Wo = (const float*)d_in[14];
  const float* gWx_f = (const float*)d_in[15];
  const float* gWh_f = (const float*)d_in[16];
  const float* gbx_f = (const float*)d_in[17];
  const float* gbh_f = (const float*)d_in[18];
  const float* gWx_b = (const float*)d_in[19];
  const float* gWh_b = (const float*)d_in[20];
  const float* gbx_b = (const float*)d_in[21];
  const float* gbh_b = (const float*)d_in[22];
  const float* fc_W = (const float*)d_in[23];
  const float* fc_b = (const float*)d_in[24];

  float* ws = (float*)d_ws;
  float* x = ws;                   // [3200,64]
  float* q = ws + 204800;          // [3200,64]
  float* kk = ws + 409600;         // [3200,64]
  float* vv = ws + 614400;         // [3200,64]
  float* av = ws + 819200;         // [3200,64]
  float* trans = ws + 1024000;     // [3200,64]
  float* xgf = ws + 1228800;       // [3200,192]
  float* xgb = ws + 1843200;       // [3200,192]
  float* gsf = ws + 2457600;       // [64,64]
  float* gsb = ws + 2461696;       // [64,64]

  k0_kgat<<<RS, 256, 0, stream>>>(u, click_seq, adj_ent, adj_rel, usr_emb,
                                  ent_emb, rel_emb, agg_W, agg_b, pos, x);
  k_rowgemm64<<<RS / 16, 128, 0, stream>>>(x, Wq, nullptr, q, 0);
  k_rowgemm64<<<RS / 16, 128, 0, stream>>>(x, Wk, nullptr, kk, 0);
  k_rowgemm64<<<RS / 16, 128, 0, stream>>>(x, Wv, nullptr, vv, 0);
  k_attn<<<Bn, 256, 0, stream>>>(q, kk, vv, av);
  k_rowgemm64<<<RS / 16, 128, 0, stream>>>(av, Wo, x, trans, 1);
  k_xg<<<dim3(RS / 16, 3), 128, 0, stream>>>(trans, gWx_f, gbx_f, xgf);
  k_xg<<<dim3(RS / 16, 3), 128, 0, stream>>>(trans, gWx_b, gbx_b, xgb);
  k_gru<<<4, 128, 0, stream>>>(xgf, gWh_f, gbh_f, 0, gsf);
  k_gru<<<4, 128, 0, stream>>>(xgb, gWh_b, gbh_b, 1, gsb);
  k_final<<<Bn, 64, 0, stream>>>(gsf, gsb, trans, fc_W, fc_b, ent_emb, v,
                                 (float*)d_out);
}